// SAGENN_80075370266803
// MI455X (gfx1250) — compile-verified
//
#include <hip/hip_runtime.h>
#include <hip/hip_bf16.h>

// ---------------------------------------------------------------------------
// SAGEConv x2 on MI455X (gfx1250, wave32).
//   agg = segment_sum(h[src], dst) * inv_deg ; out = agg@Wl + b + h@Wr
// Scatter-bound problem (~1.6 GB of edge traffic, vs 6.6 GFLOP of GEMM).
// GEMM uses V_WMMA_F32_16X16X4_F32 (full fp32 precision, matches reference).
// ---------------------------------------------------------------------------

typedef __attribute__((ext_vector_type(2))) float v2f;
typedef __attribute__((ext_vector_type(8))) float v8f;

#define NN 50000
#define NE 800000
#define DD 128

// ---------------------------------------------------------------- zero fill
__global__ void sage_zero_f32(float* __restrict__ p, int n) {
    int i = blockIdx.x * blockDim.x + threadIdx.x;
    if (i < n) p[i] = 0.0f;
}

// ------------------------------------------------------------- in-degrees
__global__ void sage_deg(const long long* __restrict__ dst,
                         float* __restrict__ deg, int E) {
    int e = blockIdx.x * blockDim.x + threadIdx.x;
    if (e < E) {
        int d = (int)dst[e];
        unsafeAtomicAdd(&deg[d], 1.0f);     // global_atomic_add_f32 (hits L2)
    }
}

__global__ void sage_invdeg(float* __restrict__ deg, int n) {
    int i = blockIdx.x * blockDim.x + threadIdx.x;
    if (i < n) deg[i] = 1.0f / fmaxf(deg[i], 1.0f);
}

// ---------------------------------------------------- edge scatter (sum agg)
// One wave32 per edge: lanes cover the 128-wide feature row (4 cols/lane).
// Loads are coalesced; atomics land in L2 (agg = 25.6MB << 192MB L2).
__global__ void sage_scatter(const float* __restrict__ feat,
                             const long long* __restrict__ src,
                             const long long* __restrict__ dst,
                             float* __restrict__ agg, int E) {
    int gw   = (blockIdx.x * blockDim.x + threadIdx.x) >> 5;  // global wave id
    int lane = threadIdx.x & 31;
    if (gw >= E) return;
    int s = (int)src[gw];
    int d = (int)dst[gw];
    const float* xs = feat + (size_t)s * DD;
    float*       ad = agg  + (size_t)d * DD;
#pragma unroll
    for (int c = 0; c < DD; c += 32)
        unsafeAtomicAdd(&ad[c + lane], xs[c + lane]);
}

// ------------------------------------------------------- fused SAGE GEMM
// Out[16x16 tile] = (inv_deg .* Agg) @ Wl  +  X @ Wr  + bias  (opt. ReLU)
// One wave per tile, 4 waves / 128-thread block.
// V_WMMA_F32_16X16X4_F32 VGPR layouts (ISA 7.12.2, 32-bit data):
//   A 16x4 : lane m (0-15) VGPR0=K0 VGPR1=K1 ; lane m+16 VGPR0=K2 VGPR1=K3
//   B 4x16 : lanes 0-15 rows K0/K1 at N=lane ; lanes 16-31 rows K2/K3
//   C 16x16: VGPR r -> (M=r, N=lane) lo-half, (M=r+8, N=lane-16) hi-half
__global__ void sage_gemm(const float* __restrict__ Agg,     // [N,D] raw sums
                          const float* __restrict__ invdeg,  // [N]
                          const float* __restrict__ X,       // [N,D] self feats
                          const float* __restrict__ Wl,      // [D,D] row-major
                          const float* __restrict__ Wr,      // [D,D] row-major
                          const float* __restrict__ bias,    // [D]
                          float* __restrict__ Out,           // [N,D]
                          int nNodes, int doRelu) {
    const int lane  = threadIdx.x & 31;
    const int wave  = threadIdx.x >> 5;
    const int tilesN = DD / 16;                       // 8 column tiles
    const int tile  = blockIdx.x * (blockDim.x >> 5) + wave;
    const int tileM = tile / tilesN;
    const int tileN = tile % tilesN;
    const int rowBase = tileM * 16;
    if (rowBase >= nNodes) return;                    // 50000 % 16 == 0 anyway

    const int m  = lane & 15;       // A row / output column within tile
    const int hi = lane >> 4;       // half-wave select
    const int kb = hi * 2;          // K sub-offset {0,2}
    const int n  = tileN * 16 + m;  // output column
    const int row = rowBase + m;    // A / X row for this lane

    const float  s    = invdeg[row];          // mean-agg scale folded into A
    const float* arow = Agg + (size_t)row * DD;
    const float* xrow = X   + (size_t)row * DD;

    v8f acc = {};

    // term 1: (inv_deg .* agg) @ Wl
#pragma unroll 4
    for (int k = 0; k < DD; k += 4) {
        v2f a, b;
        a.x = arow[k + kb]     * s;
        a.y = arow[k + kb + 1] * s;
        b.x = Wl[(size_t)(k + kb)     * DD + n];
        b.y = Wl[(size_t)(k + kb + 1) * DD + n];
        acc = __builtin_amdgcn_wmma_f32_16x16x4_f32(
                  false, a, false, b, (short)0, acc, false, false);
    }
    // term 2: x @ Wr (same accumulator)
#pragma unroll 4
    for (int k = 0; k < DD; k += 4) {
        v2f a, b;
        a.x = xrow[k + kb];
        a.y = xrow[k + kb + 1];
        b.x = Wr[(size_t)(k + kb)     * DD + n];
        b.y = Wr[(size_t)(k + kb + 1) * DD + n];
        acc = __builtin_amdgcn_wmma_f32_16x16x4_f32(
                  false, a, false, b, (short)0, acc, false, false);
    }

    // epilogue: bias (+ReLU), scatter accumulator per C/D layout
    const float bn = bias[n];
#pragma unroll
    for (int r = 0; r < 8; ++r) {
        int   orow = rowBase + r + 8 * hi;
        float v    = acc[r] + bn;
        if (doRelu) v = fmaxf(v, 0.0f);
        Out[(size_t)orow * DD + n] = v;
    }
}

// ---------------------------------------------------------------------------
extern "C" void kernel_launch(void* const* d_in, const int* in_sizes, int n_in,
                              void* d_out, int out_size, void* d_ws, size_t ws_size,
                              hipStream_t stream) {
    const float*     x   = (const float*)d_in[0];
    const long long* ei  = (const long long*)d_in[1];   // int64 [2, E]
    const float*     W1l = (const float*)d_in[2];
    const float*     b1  = (const float*)d_in[3];
    const float*     W1r = (const float*)d_in[4];
    const float*     W2l = (const float*)d_in[5];
    const float*     b2  = (const float*)d_in[6];
    const float*     W2r = (const float*)d_in[7];
    float*           out = (float*)d_out;

    const long long* src = ei;        // row 0
    const long long* dst = ei + NE;   // row 1

    // workspace layout (floats): deg[NN] | agg[NN*DD] | h[NN*DD]  (~51.5 MB)
    float* deg = (float*)d_ws;
    float* agg = deg + 51200;                    // padded past NN
    float* h   = agg + (size_t)NN * DD;

    const int T = 256;
    const int nAgg     = NN * DD;                          // 6,400,000
    const int scatterB = (NE * 32 + T - 1) / T;            // wave per edge
    const int gemmB    = (NN / 16) * (DD / 16) / 4;        // 4 tiles/block

    // degree + inverse
    sage_zero_f32<<<(NN + T - 1) / T, T, 0, stream>>>(deg, NN);
    sage_deg<<<(NE + T - 1) / T, T, 0, stream>>>(dst, deg, NE);
    sage_invdeg<<<(NN + T - 1) / T, T, 0, stream>>>(deg, NN);

    // layer 1
    sage_zero_f32<<<(nAgg + T - 1) / T, T, 0, stream>>>(agg, nAgg);
    sage_scatter<<<scatterB, T, 0, stream>>>(x, src, dst, agg, NE);
    sage_gemm<<<gemmB, 128, 0, stream>>>(agg, deg, x, W1l, W1r, b1, h, NN, 1);

    // layer 2
    sage_zero_f32<<<(nAgg + T - 1) / T, T, 0, stream>>>(agg, nAgg);
    sage_scatter<<<scatterB, T, 0, stream>>>(h, src, dst, agg, NE);
    sage_gemm<<<gemmB, 128, 0, stream>>>(agg, deg, h, W2l, W2r, b2, out, NN, 0);
}